// GAT_22119081574526
// MI455X (gfx1250) — compile-verified
//
#include <hip/hip_runtime.h>

// ---------------------------------------------------------------------------
// Two-layer GAT forward for MI455X (gfx1250, wave32).
//  - GEMMs via V_WMMA_F32_16X16X4_F32 (exact f32 matrix path)
//  - edge softmax via ordered-uint atomicMax + f32 atomicAdd (L2-resident)
// ---------------------------------------------------------------------------

typedef float v2f __attribute__((ext_vector_type(2)));
typedef float v8f __attribute__((ext_vector_type(8)));

// ---------------- WMMA f32 GEMM: C[M,NCOL] = A[M,K] * B[K,NCOL] -------------
// One wave computes a 16-row x (NT*16)-col tile. M must be multiple of 16.
template <int NT, int K, int NCOL>
__global__ __launch_bounds__(256) void gemm_wmma_f32(const float* __restrict__ A,
                                                     const float* __restrict__ B,
                                                     float* __restrict__ C,
                                                     int mtiles) {
  int wave = (int)((blockIdx.x * blockDim.x + threadIdx.x) >> 5);
  int lane = (int)(threadIdx.x & 31);
  if (wave < mtiles) {
    const int lane15 = lane & 15;
    const int khalf  = (lane >> 4) * 2;        // 0 for lanes 0-15, 2 for 16-31
    const long rowbase = (long)wave * 16;

    v8f acc[NT];
#pragma unroll
    for (int t = 0; t < NT; ++t) acc[t] = (v8f){0.f,0.f,0.f,0.f,0.f,0.f,0.f,0.f};

    const float* arow = A + (rowbase + lane15) * K + khalf;
#pragma unroll 4
    for (int kb = 0; kb < K; kb += 4) {
      v2f a;
      a.x = arow[kb];
      a.y = arow[kb + 1];
#pragma unroll
      for (int t = 0; t < NT; ++t) {
        const float* bp = B + (size_t)(kb + khalf) * NCOL + t * 16 + lane15;
        v2f b;
        b.x = bp[0];
        b.y = bp[NCOL];
        acc[t] = __builtin_amdgcn_wmma_f32_16x16x4_f32(
            /*neg_a=*/false, a, /*neg_b=*/false, b,
            /*c_mod=*/(short)0, acc[t], /*reuse_a=*/false, /*reuse_b=*/false);
      }
    }
    const int rowoff = (lane >> 4) * 8;        // lanes 16-31 hold M = j+8
#pragma unroll
    for (int t = 0; t < NT; ++t) {
#pragma unroll
      for (int j = 0; j < 8; ++j) {
        C[(rowbase + rowoff + j) * NCOL + t * 16 + lane15] = acc[t][j];
      }
    }
  }
}

// ---------------- per-node attention scalars: a = <h[n,head,:], att[head,:]> -
template <int H, int C>
__global__ void att_k(const float* __restrict__ h, const float* __restrict__ att_s,
                      const float* __restrict__ att_d, float* __restrict__ a_s,
                      float* __restrict__ a_d, int n) {
  int i = blockIdx.x * blockDim.x + threadIdx.x;
  if (i >= n * H) return;
  int node = i / H, head = i % H;
  const float* hp = h + (long)node * (H * C) + head * C;
  float s = 0.f, d = 0.f;
#pragma unroll
  for (int c = 0; c < C; ++c) {
    float v = hp[c];
    s += v * att_s[head * C + c];
    d += v * att_d[head * C + c];
  }
  a_s[i] = s;
  a_d[i] = d;
}

// ---------------- ordered-uint encoding for float atomic max ----------------
__device__ __forceinline__ unsigned ordenc(float f) {
  unsigned b = __float_as_uint(f);
  return (b & 0x80000000u) ? ~b : (b | 0x80000000u);
}
__device__ __forceinline__ float orddec(unsigned u) {
  unsigned b = (u & 0x80000000u) ? (u & 0x7fffffffu) : ~u;
  return __uint_as_float(b);
}

__device__ __forceinline__ void edge_ids(const int* __restrict__ ei, int E, int i,
                                         int& s, int& d) {
  if (i < E) { s = ei[i]; d = ei[E + i]; }
  else       { s = i - E; d = i - E; }     // self loop
}

// ---------------- pass 1: segment max of leaky-relu(e) over dst -------------
template <int H>
__global__ void edge_max_k(const int* __restrict__ ei, int E, int n,
                           const float* __restrict__ a_s, const float* __restrict__ a_d,
                           unsigned* __restrict__ om) {
  int i = blockIdx.x * blockDim.x + threadIdx.x;
  if (i >= E + n) return;
  int s, d; edge_ids(ei, E, i, s, d);
#pragma unroll
  for (int h = 0; h < H; ++h) {
    float e = a_s[(long)s * H + h] + a_d[(long)d * H + h];
    e = (e >= 0.f) ? e : 0.2f * e;
    atomicMax(&om[(long)d * H + h], ordenc(e));
  }
}

// ---------------- pass 2: e_exp + segment sum of denominators ---------------
template <int H>
__global__ void edge_exp_k(const int* __restrict__ ei, int E, int n,
                           const float* __restrict__ a_s, const float* __restrict__ a_d,
                           const unsigned* __restrict__ om, float* __restrict__ den,
                           float* __restrict__ eexp) {
  int i = blockIdx.x * blockDim.x + threadIdx.x;
  if (i >= E + n) return;
  int s, d; edge_ids(ei, E, i, s, d);
#pragma unroll
  for (int h = 0; h < H; ++h) {
    float e = a_s[(long)s * H + h] + a_d[(long)d * H + h];
    e = (e >= 0.f) ? e : 0.2f * e;
    float ex = __expf(e - orddec(om[(long)d * H + h]));
    eexp[(long)i * H + h] = ex;
    atomicAdd(&den[(long)d * H + h], ex);
  }
}

// ---------------- pass 3 (layer 1): scatter 128-f32 messages, wave/edge -----
__global__ __launch_bounds__(256) void scatter1_k(const int* __restrict__ ei, int E, int n,
                                                  const float* __restrict__ h,
                                                  const float* __restrict__ eexp,
                                                  const float* __restrict__ den,
                                                  float* __restrict__ out) {
  int wv   = (int)((blockIdx.x * blockDim.x + threadIdx.x) >> 5);  // edge id
  int lane = (int)(threadIdx.x & 31);
  if (wv >= E + n) return;
  int s, d; edge_ids(ei, E, wv, s, d);
  int head = lane >> 3;  // cols [4*lane,4*lane+3] all fall in head = lane/8
  float alpha = eexp[(long)wv * 4 + head] / den[(long)d * 4 + head];
  const float4 hv = *(const float4*)(h + (long)s * 128 + lane * 4);
  float* op = out + (long)d * 128 + lane * 4;
  atomicAdd(op + 0, hv.x * alpha);
  atomicAdd(op + 1, hv.y * alpha);
  atomicAdd(op + 2, hv.z * alpha);
  atomicAdd(op + 3, hv.w * alpha);
}

// ---------------- pass 3 (layer 2): 16-f32 messages, 16 lanes/edge ----------
__global__ void scatter2_k(const int* __restrict__ ei, int E, int n,
                           const float* __restrict__ h2, const float* __restrict__ eexp,
                           const float* __restrict__ den, float* __restrict__ out) {
  long idx = (long)blockIdx.x * blockDim.x + threadIdx.x;
  long tot = (long)(E + n) * 16;
  if (idx >= tot) return;
  int e = (int)(idx >> 4);
  int c = (int)(idx & 15);
  int s, d; edge_ids(ei, E, e, s, d);
  float alpha = eexp[e] / den[d];
  atomicAdd(&out[(long)d * 16 + c], h2[(long)s * 16 + c] * alpha);
}

// ---------------- utility fills / activations -------------------------------
__global__ void fill_u32_k(unsigned* __restrict__ p, unsigned v, long cnt) {
  long i = (long)blockIdx.x * blockDim.x + threadIdx.x;
  if (i < cnt) p[i] = v;
}
template <int C>  // C power of two
__global__ void init_bias_k(float* __restrict__ p, const float* __restrict__ b, long cnt) {
  long i = (long)blockIdx.x * blockDim.x + threadIdx.x;
  if (i < cnt) p[i] = b[i & (C - 1)];
}
__global__ void elu_k(float* __restrict__ p, long cnt) {
  long i = (long)blockIdx.x * blockDim.x + threadIdx.x;
  if (i < cnt) {
    float x = p[i];
    p[i] = (x > 0.f) ? x : expm1f(x);
  }
}

// ---------------------------------------------------------------------------
extern "C" void kernel_launch(void* const* d_in, const int* in_sizes, int n_in,
                              void* d_out, int out_size, void* d_ws, size_t ws_size,
                              hipStream_t stream) {
  const float* x    = (const float*)d_in[0];
  const int*   ei   = (const int*)d_in[1];
  const float* W1   = (const float*)d_in[2];
  const float* as1w = (const float*)d_in[3];
  const float* ad1w = (const float*)d_in[4];
  const float* b1   = (const float*)d_in[5];
  const float* W2   = (const float*)d_in[6];
  const float* as2w = (const float*)d_in[7];
  const float* ad2w = (const float*)d_in[8];
  const float* b2   = (const float*)d_in[9];
  float* out = (float*)d_out;

  const int n   = in_sizes[0] / 128;  // 100000
  const int E   = in_sizes[1] / 2;    // 1600000
  const int tot = E + n;              // edges + self loops

  // ---- workspace layout (f32 units) ----
  float* ws   = (float*)d_ws;
  float*    h1   = ws;                                  // n*128
  float*    out1 = h1 + (size_t)n * 128;                // n*128
  float*    as1  = out1 + (size_t)n * 128;              // n*4
  float*    ad1  = as1 + (size_t)n * 4;                 // n*4
  unsigned* om1  = (unsigned*)(ad1 + (size_t)n * 4);    // n*4
  float*    den1 = (float*)(om1 + (size_t)n * 4);       // n*4
  float*    eexp = den1 + (size_t)n * 4;                // tot*4 (shared both layers)
  // layer-2 small buffers overlay h1 (h1 dead after layer-1 scatter)
  float*    h2   = h1;                                  // n*16
  float*    as2  = h1 + (size_t)n * 16;                 // n
  float*    ad2  = as2 + n;                             // n
  unsigned* om2  = (unsigned*)(ad2 + n);                // n
  float*    den2 = (float*)(om2 + n);                   // n

  const unsigned ORD_NEG_INF = 0x007FFFFFu;  // ordenc(-inf)
  const int B = 256;
  const int mtiles = (n + 15) / 16;          // 6250 (exact)

  // ================= Layer 1: GATConv(128 -> 4x32) =================
  gemm_wmma_f32<8, 128, 128><<<(mtiles + 7) / 8, B, 0, stream>>>(x, W1, h1, mtiles);
  att_k<4, 32><<<(n * 4 + B - 1) / B, B, 0, stream>>>(h1, as1w, ad1w, as1, ad1, n);
  fill_u32_k<<<(int)(((long)n * 4 + B - 1) / B), B, 0, stream>>>(om1, ORD_NEG_INF, (long)n * 4);
  fill_u32_k<<<(int)(((long)n * 4 + B - 1) / B), B, 0, stream>>>((unsigned*)den1, 0u, (long)n * 4);
  init_bias_k<128><<<(int)(((long)n * 128 + B - 1) / B), B, 0, stream>>>(out1, b1, (long)n * 128);
  edge_max_k<4><<<(tot + B - 1) / B, B, 0, stream>>>(ei, E, n, as1, ad1, om1);
  edge_exp_k<4><<<(tot + B - 1) / B, B, 0, stream>>>(ei, E, n, as1, ad1, om1, den1, eexp);
  scatter1_k<<<(int)(((long)tot * 32 + B - 1) / B), B, 0, stream>>>(ei, E, n, h1, eexp, den1, out1);
  elu_k<<<(int)(((long)n * 128 + B - 1) / B), B, 0, stream>>>(out1, (long)n * 128);

  // ================= Layer 2: GATConv(128 -> 1x16) =================
  gemm_wmma_f32<1, 128, 16><<<(mtiles + 7) / 8, B, 0, stream>>>(out1, W2, h2, mtiles);
  att_k<1, 16><<<(n + B - 1) / B, B, 0, stream>>>(h2, as2w, ad2w, as2, ad2, n);
  fill_u32_k<<<(n + B - 1) / B, B, 0, stream>>>(om2, ORD_NEG_INF, (long)n);
  fill_u32_k<<<(n + B - 1) / B, B, 0, stream>>>((unsigned*)den2, 0u, (long)n);
  init_bias_k<16><<<(int)(((long)n * 16 + B - 1) / B), B, 0, stream>>>(out, b2, (long)n * 16);
  edge_max_k<1><<<(tot + B - 1) / B, B, 0, stream>>>(ei, E, n, as2, ad2, om2);
  edge_exp_k<1><<<(tot + B - 1) / B, B, 0, stream>>>(ei, E, n, as2, ad2, om2, den2, eexp);
  scatter2_k<<<(int)(((long)tot * 16 + B - 1) / B), B, 0, stream>>>(ei, E, n, h2, eexp, den2, out);
}